// PCC_Loss_90907277787183
// MI455X (gfx1250) — compile-verified
//
#include <hip/hip_runtime.h>
#include <hip/hip_bf16.h>

// PCC loss, MI455X (gfx1250, wave32).
// Memory-bound: 256 MiB streamed once -> ~11.5 us floor @ 23.3 TB/s.
// Stage 1: streaming 5-moment accumulation, NT b128 loads, 2048 blocks x 8 waves.
// Stage 2: per-batch chunk reduction via V_WMMA_F32_16X16X4_F32 (ones x partials),
//          then per-channel PCC + wave reduction.
// Stage 3: mean over batches -> d_out[0].

typedef float v2f __attribute__((ext_vector_type(2)));
typedef float v4f __attribute__((ext_vector_type(4)));
typedef float v8f __attribute__((ext_vector_type(8)));

constexpr int Bn = 64;
constexpr int Tn = 8192;
constexpr int Cn = 64;
constexpr int NCHUNK = 32;            // T-chunks per batch
constexpr int CT = Tn / NCHUNK;       // 256 timesteps per chunk
constexpr int NCOL = 5 * Cn;          // 320 partial columns: [sx|sy|sxy|sxx|syy] x 64 ch

// ---------------- Stage 1: streaming moment accumulation ----------------
// grid = (NCHUNK, B), block = 256.  Thread (row = tid>>4, c4 = tid&15) loads
// float4 of channels [4*c4, 4*c4+4) for timestep t = chunk*CT + row + 16*i.
// A wave32 covers 2 adjacent t-rows = 512 contiguous bytes per load.
__global__ __launch_bounds__(256) void pcc_stage1(
    const float* __restrict__ x, const float* __restrict__ y,
    const int* __restrict__ lens, float* __restrict__ partial) {
  const int chunk = blockIdx.x;
  const int b     = blockIdx.y;
  const int tid   = threadIdx.x;
  const int c4    = tid & 15;
  const int row   = tid >> 4;
  const int L     = lens[b];
  const int t0    = chunk * CT;

  const float* xb = x + (size_t)b * Tn * Cn + (size_t)c4 * 4;
  const float* yb = y + (size_t)b * Tn * Cn + (size_t)c4 * 4;

  v4f sx  = {0.f, 0.f, 0.f, 0.f};
  v4f sy  = {0.f, 0.f, 0.f, 0.f};
  v4f sxy = {0.f, 0.f, 0.f, 0.f};
  v4f sxx = {0.f, 0.f, 0.f, 0.f};
  v4f syy = {0.f, 0.f, 0.f, 0.f};

#pragma unroll 4
  for (int i = 0; i < CT; i += 16) {
    const int t = t0 + row + i;
    const v4f xv = __builtin_nontemporal_load((const v4f*)(xb + (size_t)t * Cn));
    const v4f yv = __builtin_nontemporal_load((const v4f*)(yb + (size_t)t * Cn));
    if (t < L) {                       // L >= T/2: only chunks >= 16 diverge
      sx  += xv;
      sy  += yv;
      sxy += xv * yv;
      sxx += xv * xv;
      syy += yv * yv;
    }
  }

  __shared__ float sm[256][21];        // pad 20->21: conflict-free banks
#pragma unroll
  for (int j = 0; j < 4; ++j) {
    sm[tid][0  + j] = sx[j];
    sm[tid][4  + j] = sy[j];
    sm[tid][8  + j] = sxy[j];
    sm[tid][12 + j] = sxx[j];
    sm[tid][16 + j] = syy[j];
  }
  __syncthreads();

  // Reduce the 16 row-groups (strides multiples of 16 keep c4 aligned).
  for (int stride = 128; stride >= 16; stride >>= 1) {
    if (tid < stride) {
#pragma unroll
      for (int s = 0; s < 20; ++s) sm[tid][s] += sm[tid + stride][s];
    }
    __syncthreads();
  }

  if (tid < 16) {
    float* dst = partial + (size_t)(b * NCHUNK + chunk) * NCOL;
#pragma unroll
    for (int s = 0; s < 5; ++s)
#pragma unroll
      for (int j = 0; j < 4; ++j)
        dst[s * Cn + tid * 4 + j] = sm[tid][s * 4 + j];   // col = s*64 + channel
  }
}

// ---------------- Stage 2: per-batch WMMA reduction + PCC ----------------
// grid = B, block = 32 (one wave, EXEC all-ones for WMMA).
// colsum[n] = sum over 32 chunks of partial[chunk][n], computed as
// ones(16x4) x B(4x16) accumulated over 8 K-steps. Column n lives in lanes
// n and n+16; which K-row each slot is doesn't matter (sum over K).
__global__ __launch_bounds__(32) void pcc_stage2(
    const float* __restrict__ partial, const int* __restrict__ lens,
    float* __restrict__ perb) {
  const int b    = blockIdx.x;
  const int lane = threadIdx.x;
  const int n16  = lane & 15;
  const int grp  = lane >> 4;          // 0: chunks k,k+2  1: chunks k+1,k+3
  const float* P = partial + (size_t)b * NCHUNK * NCOL;

  __shared__ float colsum[NCOL];
  const v2f ones = {1.0f, 1.0f};

  for (int tile = 0; tile < NCOL / 16; ++tile) {
    const int col = tile * 16 + n16;
    v8f acc = {0.f, 0.f, 0.f, 0.f, 0.f, 0.f, 0.f, 0.f};
#pragma unroll
    for (int ks = 0; ks < NCHUNK; ks += 4) {
      v2f bv;
      bv.x = P[(size_t)(ks + grp)     * NCOL + col];
      bv.y = P[(size_t)(ks + 2 + grp) * NCOL + col];
      // D = ones(16x4) * bv(4x16) + acc  ->  every D row = column sums
      acc = __builtin_amdgcn_wmma_f32_16x16x4_f32(
          /*neg_a=*/false, ones, /*neg_b=*/false, bv,
          /*c_mod=*/(short)0, acc, /*reuse_a=*/false, /*reuse_b=*/false);
    }
    if (lane < 16) colsum[col] = acc[0];   // D[0, n=lane]
  }
  __syncthreads();

  const float n = (float)lens[b];
  float psum = 0.f;
#pragma unroll
  for (int k = 0; k < 2; ++k) {
    const int ch = lane + k * 32;
    const float sx  = colsum[0 * Cn + ch];
    const float sy  = colsum[1 * Cn + ch];
    const float sxy = colsum[2 * Cn + ch];
    const float sxx = colsum[3 * Cn + ch];
    const float syy = colsum[4 * Cn + ch];
    const float cov = sxy - sx * sy / n;
    const float vx  = sxx - sx * sx / n;
    const float vy  = syy - sy * sy / n;
    psum += cov / sqrtf(vx * vy);
  }
  for (int off = 16; off; off >>= 1) psum += __shfl_xor(psum, off, 32);
  if (lane == 0) perb[b] = 1.0f - psum * (1.0f / (float)Cn);
}

// ---------------- Stage 3: mean over batches ----------------
__global__ __launch_bounds__(32) void pcc_stage3(
    const float* __restrict__ perb, float* __restrict__ out) {
  const int lane = threadIdx.x;
  float v = perb[lane] + perb[lane + 32];
  for (int off = 16; off; off >>= 1) v += __shfl_xor(v, off, 32);
  if (lane == 0) out[0] = v * (1.0f / (float)Bn);
}

extern "C" void kernel_launch(void* const* d_in, const int* in_sizes, int n_in,
                              void* d_out, int out_size, void* d_ws, size_t ws_size,
                              hipStream_t stream) {
  (void)in_sizes; (void)n_in; (void)out_size; (void)ws_size;
  const float* x   = (const float*)d_in[0];
  const float* y   = (const float*)d_in[1];
  const int* lens  = (const int*)d_in[2];

  float* partial = (float*)d_ws;                       // 64*32*320 f32 = 2.62 MB
  float* perb    = partial + (size_t)Bn * NCHUNK * NCOL;

  dim3 g1(NCHUNK, Bn);
  pcc_stage1<<<g1, 256, 0, stream>>>(x, y, lens, partial);
  pcc_stage2<<<Bn, 32, 0, stream>>>(partial, lens, perb);
  pcc_stage3<<<1, 32, 0, stream>>>(perb, (float*)d_out);
}